// PositionalEncoder_54331336294435
// MI455X (gfx1250) — compile-verified
//
#include <hip/hip_runtime.h>

// out[b,s,d] = x[b,s,d] + pe[s,d]
// pe[s,d] = (d even) ? sin(s * 10000^(-d/1024)) : cos(s * 10000^(-d/1024))
//
// angle[s,d] = pos[s] * invfreq[d] is RANK-1 -> computed per 16x16 tile on the
// matrix pipe with V_WMMA_F32_16X16X4_F32 (outer product, K=0 used, K=1..3 zero).
// Trick: D = A*B + C with A pre-scaled by 1/(2pi) and C = 0.25 on odd-d lanes
// (cos(x) = sin(x + pi/2); v_sin_f32 input is in revolutions), so ALL lanes run
// a uniform v_sin_f32 -- no divergent sin/cos branching at all.
// pe cached in LDS and reused across batch=4; global traffic is pure b128 NT streams.

typedef float v2f __attribute__((ext_vector_type(2)));
typedef float v4f __attribute__((ext_vector_type(4)));
typedef float v8f __attribute__((ext_vector_type(8)));

#define S_TOKENS 8192
#define N_DIM    1024
#define BATCH    4
#define TILE_S   16
#define TILE_D   256
#define LDS_STRIDE (TILE_D + 4)   // 260: rows 8 apart -> banks shifted by 32, no conflict

__global__ __launch_bounds__(256)
void pe_add_kernel(const float* __restrict__ x, float* __restrict__ out) {
    __shared__ float pe_s[TILE_S * LDS_STRIDE];

    const int tid  = threadIdx.x;
    const int lane = tid & 31;
    const int wave = tid >> 5;
    const int d0   = blockIdx.x * TILE_D;   // global d tile base (multiple of 256)
    const int s0   = blockIdx.y * TILE_S;   // global s tile base

    // ---- Phase 1: rank-1 angle tiles on the WMMA pipe -----------------------
    // A (16x4): lanes 0-15 hold M=lane, VGPR0=K0; A[m,0] = pos[m]/(2pi), rest 0.
    // B (4x16): K=0 row lives in VGPR0 lanes 0-15; B[0,n] = invfreq[d], rest 0.
    // C (16x16): 0.25 on odd-n lanes (phase shift pi/2 in revolutions) -> cos.
    const float LOG2_10000 = 13.2877123795494f;
    const float INV_2PI    = 0.15915494309189535f;
    const float posv  = (lane < 16) ? (float)(s0 + lane) * INV_2PI : 0.0f;
    const float phase = (lane & 1) ? 0.25f : 0.0f;   // n parity == lane parity

    #pragma unroll
    for (int t = 0; t < 2; ++t) {
        const int dg0 = d0 + wave * 32 + t * 16;   // tile's global d base (even)
        float invf = 0.0f;
        if (lane < 16) {
            float dd = (float)(dg0 + lane);
            invf = __builtin_amdgcn_exp2f(-(dd * (1.0f / (float)N_DIM)) * LOG2_10000);
        }
        v2f a; a.x = posv; a.y = 0.0f;
        v2f b; b.x = invf; b.y = 0.0f;
        v8f c = {phase, phase, phase, phase, phase, phase, phase, phase};
        // D = A x B + C : angle tile in revolutions (+ pi/2 phase on odd d)
        c = __builtin_amdgcn_wmma_f32_16x16x4_f32(
                /*neg_a=*/false, a, /*neg_b=*/false, b,
                /*c_mod=*/(short)0, c, /*reuse_a=*/false, /*reuse_b=*/false);

        // D layout: VGPR r -> lanes 0-15: (m=r, n=lane); lanes 16-31: (m=r+8, n=lane-16)
        const int n     = lane & 15;
        const int mbase = (lane >> 4) * 8;
        const int dl    = wave * 32 + t * 16 + n;
        #pragma unroll
        for (int r = 0; r < 8; ++r) {
            // uniform across the wave: v_sin_f32 (input already in revolutions)
            pe_s[(mbase + r) * LDS_STRIDE + dl] = __builtin_amdgcn_sinf(c[r]);
        }
    }
    __syncthreads();

    // ---- Phase 2: coalesced b128 NT streaming add, pe reused over batch -----
    // Tile = 16 rows x 64 float4-cols = 1024 float4 slots; 256 threads x 16 iters.
    #pragma unroll 4
    for (int i = 0; i < 16; ++i) {
        const int f   = i * 256 + tid;
        const int row = f >> 6;          // 0..63
        const int c4  = f & 63;          // float4 column
        const int dl  = c4 * 4;
        const v4f pe4 = *(const v4f*)&pe_s[row * LDS_STRIDE + dl];
        const long base = (long)(s0 + row) * N_DIM + (long)(d0 + dl);
        #pragma unroll
        for (int bb = 0; bb < BATCH; ++bb) {
            const long idx = (long)bb * ((long)S_TOKENS * N_DIM) + base;
            v4f v = __builtin_nontemporal_load((const v4f*)(x + idx));
            v += pe4;
            __builtin_nontemporal_store(v, (v4f*)(out + idx));
        }
    }
}

extern "C" void kernel_launch(void* const* d_in, const int* in_sizes, int n_in,
                              void* d_out, int out_size, void* d_ws, size_t ws_size,
                              hipStream_t stream) {
    const float* x  = (const float*)d_in[0];
    float*      out = (float*)d_out;
    dim3 grid(N_DIM / TILE_D, S_TOKENS / TILE_S, 1);   // 4 x 512 = 2048 blocks
    pe_add_kernel<<<grid, dim3(256), 0, stream>>>(x, out);
}